// ProposalAssignment_20169166422206
// MI455X (gfx1250) — compile-verified
//
#include <hip/hip_runtime.h>
#include <hip/hip_bf16.h>

#define BB 32
#define NN 8192
#define GG 128
#define MM (NN + GG)      // 8320
#define NPAD 16384        // next pow2 >= MM for bitonic sort
#define KOUT 512          // BATCH_SIZE_PER_IM
#define MAXFG 128

// Use gfx1250 async global->LDS copies when the toolchain exposes them.
// (Host compilation pass and older toolchains take the fallback branch.)
#if defined(__AMDGCN__) && __has_builtin(__builtin_amdgcn_global_load_async_to_lds_b64) && \
    __has_builtin(__builtin_amdgcn_s_wait_asynccnt)
#define USE_ASYNC_LDS 1
#else
#define USE_ASYNC_LDS 0
#endif

#if USE_ASYNC_LDS
// Builtin signature (from clang-22 diagnostic): first param is
// `v2i __attribute__((address_space(1)))*` (prints as `__device__`),
// second is the LDS-side pointer. b64 moves 2 dwords per lane.
typedef int v2i32 __attribute__((vector_size(2 * sizeof(int))));
typedef __attribute__((address_space(1))) v2i32 gas_v2i32;  // global (AS1)
typedef __attribute__((address_space(3))) v2i32 las_v2i32;  // LDS (AS3)
#endif

// ---------------------------------------------------------------------------
// Kernel 1: per (b,m) max-IoU + argmax over the 128 GT boxes.
// gt_boxes[b] (2 KB) staged into LDS with gfx1250 async-to-LDS DMA.
// ---------------------------------------------------------------------------
__global__ __launch_bounds__(256) void iou_argmax_kernel(
    const float* __restrict__ boxes,     // [B, N, 4]
    const float* __restrict__ gt_boxes,  // [B, G, 4]
    float* __restrict__ max_ov,          // [B, M]
    int* __restrict__ p2l) {             // [B, M]
  __shared__ float sg[GG * 4];  // 512 floats = 2 KB

  const int b = blockIdx.y;
  const int t = threadIdx.x;            // 0..255
  const int m = blockIdx.x * 256 + t;
  const float* gtb = gt_boxes + (size_t)b * GG * 4;

  // Stage gt boxes: each of the 256 threads moves 8 bytes (512 floats total).
#if USE_ASYNC_LDS
  __builtin_amdgcn_global_load_async_to_lds_b64(
      (gas_v2i32*)(gtb + 2 * t),
      (las_v2i32*)(&sg[2 * t]),
      0, 0);
  __builtin_amdgcn_s_wait_asynccnt(0);
  __syncthreads();
#else
  sg[2 * t] = gtb[2 * t];
  sg[2 * t + 1] = gtb[2 * t + 1];
  __syncthreads();
#endif

  if (m >= MM) return;

  const float* bp = (m < NN) ? (boxes + ((size_t)b * NN + m) * 4)
                             : (gt_boxes + ((size_t)b * GG + (m - NN)) * 4);
  __builtin_prefetch(bp, 0, 0);  // -> global_prefetch_b8 on gfx1250
  const float4 bb = *(const float4*)bp;
  const float by0 = bb.x, bx0 = bb.y, by1 = bb.z, bx1 = bb.w;
  const float barea = (by1 - by0) * (bx1 - bx0);

  float best = -3.0f;
  int bestg = 0;
#pragma unroll 4
  for (int g = 0; g < GG; ++g) {
    const float gy0 = sg[4 * g + 0];
    const float gx0 = sg[4 * g + 1];
    const float gy1 = sg[4 * g + 2];
    const float gx1 = sg[4 * g + 3];
    const float iy0 = fmaxf(by0, gy0);
    const float ix0 = fmaxf(bx0, gx0);
    const float iy1 = fminf(by1, gy1);
    const float ix1 = fminf(bx1, gx1);
    const float ia = fmaxf(iy1 - iy0, 0.0f) * fmaxf(ix1 - ix0, 0.0f);
    const float ga = (gy1 - gy0) * (gx1 - gx0);
    float iou = ia / (barea + ga - ia);
    if (iy0 < 0.0f && ix0 < 0.0f) iou = -1.0f;  // padded-gt convention
    if (iou > best) { best = iou; bestg = g; }  // strict > keeps first argmax
  }
  max_ov[(size_t)b * MM + m] = best;
  p2l[(size_t)b * MM + m] = bestg;
}

// ---------------------------------------------------------------------------
// Kernel 2: one block per image. Single descending bitonic sort of packed
// (combined, tie-broken-by-index) keys replaces both top_k calls:
//   fg  = sorted[0 : min(npos,128)]        (all pos rank above all neg)
//   bg  = sorted[npos : ...]               (skip positives ranked >= 128)
// Then 512 threads gather all four outputs.
// ---------------------------------------------------------------------------
__global__ __launch_bounds__(1024) void sample_gather_kernel(
    const float* __restrict__ boxes,
    const float* __restrict__ gt_boxes,
    const int* __restrict__ gt_labels,   // [B, G]
    const float* __restrict__ noise,     // [B, M]
    const float* __restrict__ max_ov,    // [B, M]
    const int* __restrict__ p2l,         // [B, M]
    float* __restrict__ out) {
  extern __shared__ unsigned long long skey[];  // NPAD entries = 128 KB
  __shared__ int s_npos;

  const int b = blockIdx.x;
  const int tid = threadIdx.x;
  const int NT = blockDim.x;

  // Build sort keys: high 32 bits = order-preserving uint of `combined`,
  // low 32 bits = 0xFFFFFFFF - idx (descending sort -> lower idx wins ties,
  // matching jax.lax.top_k tie-breaking).
  for (int i = tid; i < NPAD; i += NT) {
    unsigned long long K = 0ull;  // padding sorts below every real key
    if (i < MM) {
      const float mo = max_ov[(size_t)b * MM + i];
      const float nz = noise[(size_t)b * MM + i];
      const bool pos = mo > 0.5f;
      const bool neg = (mo >= 0.0f) && (mo < 0.5f);
      const float comb = pos ? (2.0f + nz) : (neg ? nz : -1.0f);
      unsigned int ku = __float_as_uint(comb);
      ku = (ku & 0x80000000u) ? ~ku : (ku | 0x80000000u);
      K = ((unsigned long long)ku << 32) |
          (unsigned long long)(0xFFFFFFFFu - (unsigned int)i);
    }
    skey[i] = K;
  }
  __syncthreads();

  // Descending bitonic sort.
  for (int k = 2; k <= NPAD; k <<= 1) {
    for (int j = k >> 1; j > 0; j >>= 1) {
      for (int i = tid; i < NPAD; i += NT) {
        const int ixj = i ^ j;
        if (ixj > i) {
          const unsigned long long a = skey[i];
          const unsigned long long c = skey[ixj];
          const bool desc = ((i & k) == 0);
          if (desc ? (a < c) : (a > c)) {
            skey[i] = c;
            skey[ixj] = a;
          }
        }
      }
      __syncthreads();
    }
  }

  // Count positives (combined >= 2.0 -> packed key >= 0xC0000000_00000000).
  if (tid == 0) s_npos = 0;
  __syncthreads();
  {
    int cnt = 0;
    const unsigned long long TH = 0xC000000000000000ull;
    for (int i = tid; i < NPAD; i += NT) cnt += (skey[i] >= TH) ? 1 : 0;
    if (cnt) atomicAdd(&s_npos, cnt);
  }
  __syncthreads();

  if (tid < KOUT) {
    const int j = tid;
    const int npos = s_npos;
    const int nf = (npos < MAXFG) ? npos : MAXFG;
    int src = (j < nf) ? j : (npos + (j - nf));
    if (src > NPAD - 1) src = NPAD - 1;
    const unsigned long long K = skey[src];
    unsigned int m = 0xFFFFFFFFu - (unsigned int)(K & 0xFFFFFFFFu);
    if (m >= MM) m = 0;  // degenerate-fill guard (unreachable for this data)

    const float* rp = (m < NN)
                          ? (boxes + ((size_t)b * NN + m) * 4)
                          : (gt_boxes + ((size_t)b * GG + (m - NN)) * 4);
    const float4 roi = *(const float4*)rp;

    const float mo = max_ov[(size_t)b * MM + m];
    const int pl = p2l[(size_t)b * MM + m];
    const bool neg = (mo >= 0.0f) && (mo < 0.5f);
    const int cls = neg ? 0 : gt_labels[(size_t)b * GG + pl];
    const int plo = neg ? 0 : pl;
    float4 bt;
    if (neg) {
      bt = make_float4(0.0f, 0.0f, 0.0f, 0.0f);
    } else {
      bt = *(const float4*)(gt_boxes + ((size_t)b * GG + pl) * 4);
    }

    // Outputs concatenated flat in return order:
    // box_t [B,512,4] | cls_t [B,512] | rois [B,512,4] | p2l [B,512]
    const size_t o_boxt = 0;
    const size_t o_cls = (size_t)BB * KOUT * 4;
    const size_t o_roi = o_cls + (size_t)BB * KOUT;
    const size_t o_p2l = o_roi + (size_t)BB * KOUT * 4;
    const size_t base4 = ((size_t)b * KOUT + j) * 4;
    const size_t base1 = (size_t)b * KOUT + j;

    *(float4*)(out + o_boxt + base4) = bt;
    out[o_cls + base1] = (float)cls;
    *(float4*)(out + o_roi + base4) = roi;
    out[o_p2l + base1] = (float)plo;
  }
}

extern "C" void kernel_launch(void* const* d_in, const int* in_sizes, int n_in,
                              void* d_out, int out_size, void* d_ws, size_t ws_size,
                              hipStream_t stream) {
  const float* boxes = (const float*)d_in[0];     // [32, 8192, 4] f32
  const float* gt_boxes = (const float*)d_in[1];  // [32, 128, 4]  f32
  const int* gt_labels = (const int*)d_in[2];     // [32, 128]     i32
  const float* noise = (const float*)d_in[3];     // [32, 8320]    f32

  float* ws_maxov = (float*)d_ws;                                  // B*M floats
  int* ws_p2l = (int*)((char*)d_ws + (size_t)BB * MM * sizeof(float));

  dim3 g1((MM + 255) / 256, BB);
  iou_argmax_kernel<<<g1, 256, 0, stream>>>(boxes, gt_boxes, ws_maxov, ws_p2l);

  const size_t shmem = (size_t)NPAD * sizeof(unsigned long long);  // 128 KB
  static bool attr_set = false;
  if (!attr_set) {  // host-side attribute config; not a stream op, capture-safe
    (void)hipFuncSetAttribute((const void*)sample_gather_kernel,
                              hipFuncAttributeMaxDynamicSharedMemorySize,
                              (int)shmem);
    attr_set = true;
  }
  sample_gather_kernel<<<BB, 1024, shmem, stream>>>(
      boxes, gt_boxes, gt_labels, noise, ws_maxov, ws_p2l, (float*)d_out);
}